// Logalike_40072044871937
// MI455X (gfx1250) — compile-verified
//
#include <hip/hip_runtime.h>
#include <math.h>

#define C_N 512
#define S_N 256
#define DIM 16
#define MSTRIDE 17     // row-major stride for M (bank-conflict padding)
#define TSTRIDE 18     // column-major stride for T/W: 72B -> 8B-aligned pairs
#define WAVES_PER_BLOCK 8

typedef float v2f __attribute__((ext_vector_type(2)));
typedef float v8f __attribute__((ext_vector_type(8)));

// Same-wave DS ops execute in order in HW; loads->stores only need compiler
// ordering. Store->load crossings (other lanes' data) get a HW wait.
__device__ __forceinline__ void compiler_fence() { asm volatile("" ::: "memory"); }
__device__ __forceinline__ void lds_fence() { asm volatile("s_wait_dscnt 0" ::: "memory"); }

// Full 16x16x16 product: 4 chained V_WMMA_F32_16X16X4_F32 (K = 4 x 4).
__device__ __forceinline__ v8f wmma_chain(const v2f a[4], const v2f b[4]) {
  v8f acc = {};
#pragma unroll
  for (int kc = 0; kc < 4; ++kc)
    acc = __builtin_amdgcn_wmma_f32_16x16x4_f32(false, a[kc], false, b[kc],
                                                (short)0, acc, false, false);
  return acc;
}

// B fragment (ISA 7.12.2): VGPR0: K=+0 (lanes0-15) / K=+2 (lanes16-31);
// VGPR1: K=+1 / K=+3; column = lane%16. Column-major storage makes each
// fragment one aligned 8-byte contiguous load.
__device__ __forceinline__ void load_bfrag_cm(const float* B, int lane, v2f b[4]) {
  const int half = lane >> 4, l16 = lane & 15;
#pragma unroll
  for (int kc = 0; kc < 4; ++kc) {
    const int bk = kc * 4 + half * 2;
    b[kc] = *(const v2f*)(B + l16 * TSTRIDE + bk);
  }
}

// A fragment from a column-major matrix: row = lane%16, K split across halves.
__device__ __forceinline__ void load_afrag_cm(const float* A, int lane, v2f a[4]) {
  const int half = lane >> 4, l16 = lane & 15;
#pragma unroll
  for (int kc = 0; kc < 4; ++kc) {
    const int ak = kc * 4 + half * 2;
    a[kc].x = A[ak * TSTRIDE + l16];
    a[kc].y = A[(ak + 1) * TSTRIDE + l16];
  }
}

// C/D accumulator (VGPR r: row r+8*half, col lane%16) -> column-major LDS:
// rows are contiguous within a column => 4 aligned 8-byte stores.
__device__ __forceinline__ void store_cd_cm(float* D, int lane, const v8f acc) {
  const int half = lane >> 4, l16 = lane & 15;
#pragma unroll
  for (int r = 0; r < 4; ++r) {
    v2f p; p.x = acc[2 * r]; p.y = acc[2 * r + 1];
    *(v2f*)(D + l16 * TSTRIDE + half * 8 + 2 * r) = p;
  }
}

__global__ __launch_bounds__(256) void logalike_expm_kernel(
    const float* __restrict__ X, const float* __restrict__ Q,
    const int* __restrict__ ch, const int* __restrict__ ip,
    float* __restrict__ partial) {
  __shared__ float Mbuf[WAVES_PER_BLOCK][DIM * MSTRIDE];   // row-major t*Q[s]
  __shared__ float Wbuf[WAVES_PER_BLOCK][DIM * TSTRIDE];   // column-major W/T
  __shared__ float wsum[WAVES_PER_BLOCK];

  const int wave = threadIdx.x >> 5;
  const int lane = threadIdx.x & 31;
  const int half = lane >> 4;
  const int l16  = lane & 15;
  const int pair = blockIdx.x * WAVES_PER_BLOCK + wave;   // c*S + s
  const int c = pair >> 8;                                // / S_N
  const int s = pair & (S_N - 1);
  const int i = ip[0];

  // t = 0.5 * arccosh(max(-<x_i, x_c>_Minkowski, 1+1e-6))  (RHO == 1)
  float inner = -X[i * 9 + 0] * X[c * 9 + 0];
#pragma unroll
  for (int k = 1; k < 9; ++k) inner += X[i * 9 + k] * X[c * 9 + k];
  const float u = fmaxf(-inner, 1.0f + 1e-6f);
  const float t = 0.5f * logf(u + sqrtf(fmaf(u, u, -1.0f)));

  float* M = Mbuf[wave];
  float* W = Wbuf[wave];
  const float* Qs = Q + s * (DIM * DIM);

  // M = t * Q[s], row-major
#pragma unroll
  for (int r = 0; r < 8; ++r) {
    const int e = lane * 8 + r;
    M[(e >> 4) * MSTRIDE + (e & 15)] = t * Qs[e];
  }
  lds_fence();

  // ||M||_inf: wave-max of per-row abs sums
  float rowsum = 0.0f;
  if (lane < 16) {
#pragma unroll
    for (int col = 0; col < 16; ++col) rowsum += fabsf(M[lane * MSTRIDE + col]);
  }
#pragma unroll
  for (int off = 16; off > 0; off >>= 1)
    rowsum = fmaxf(rowsum, __shfl_xor(rowsum, off, 32));

  int j = 0;
  float nrm = rowsum;
  while (nrm > 0.5f && j < 24) { nrm *= 0.5f; ++j; }
  const float sc = __builtin_ldexpf(1.0f, -j);

  // Ms A-fragments live in registers for the whole Taylor phase.
  v2f am[4];
#pragma unroll
  for (int kc = 0; kc < 4; ++kc) {
    const int ak = kc * 4 + half * 2;
    am[kc].x = M[l16 * MSTRIDE + ak] * sc;
    am[kc].y = M[l16 * MSTRIDE + ak + 1] * sc;
  }

  // Seed: W = Ms (C/D-layout values per lane), S = I + Ms.  Identity is
  // added exactly once here -> no per-iteration cmp/cndmask.
  v8f S;
#pragma unroll
  for (int r = 0; r < 8; ++r) {
    const int row = r + 8 * half;
    const float w = M[row * MSTRIDE + l16] * sc;
    S[r] = w + ((row == l16) ? 1.0f : 0.0f);
  }
  store_cd_cm(W, lane, S);                 // stores I+Ms, overwritten with Ms below
  // Correction: W must hold Ms, not I+Ms. Store Ms values:
#pragma unroll
  for (int r = 0; r < 4; ++r) {
    v2f p;
    p.x = S[2 * r]     - (((2 * r     + 8 * half) == l16) ? 1.0f : 0.0f);
    p.y = S[2 * r + 1] - (((2 * r + 1 + 8 * half) == l16) ? 1.0f : 0.0f);
    *(v2f*)(W + l16 * TSTRIDE + half * 8 + 2 * r) = p;
  }
  lds_fence();

  // Direct Taylor sum: W_k = Ms * W_{k-1};  S += W_k / k!,  k = 2..8.
  // 1/k! are compile-time constants; accumulation is register-only.
#pragma unroll
  for (int k = 2; k <= 8; ++k) {
    const float invfact =
        (k == 2) ? 0.5f : (k == 3) ? (1.0f / 6.0f) : (k == 4) ? (1.0f / 24.0f)
      : (k == 5) ? (1.0f / 120.0f) : (k == 6) ? (1.0f / 720.0f)
      : (k == 7) ? (1.0f / 5040.0f) : (1.0f / 40320.0f);
    v2f bw[4];
    load_bfrag_cm(W, lane, bw);
    const v8f acc = wmma_chain(am, bw);
#pragma unroll
    for (int r = 0; r < 8; ++r) S[r] = fmaf(acc[r], invfact, S[r]);
    if (k < 8) {                       // last power is never read back
      compiler_fence();
      store_cd_cm(W, lane, acc);
      lds_fence();
    }
  }

  // T = S -> LDS (column-major), then repeated squaring: T = T*T, j times.
  compiler_fence();
  store_cd_cm(W, lane, S);
  lds_fence();

  for (int q = 0; q < j; ++q) {        // j wave-uniform -> EXEC all ones
    v2f at[4], bt[4];
    load_afrag_cm(W, lane, at);
    load_bfrag_cm(W, lane, bt);
    const v8f acc = wmma_chain(at, bt);
    compiler_fence();
    store_cd_cm(W, lane, acc);
    lds_fence();
  }

  // Gather P entries (column-major: P[row,col] at col*TSTRIDE + row)
  const int si = ch[i * S_N + s];
  const int sj = ch[c * S_N + s];
  const float p0_si = W[si * TSTRIDE + 0];
  const float p0_sj = W[sj * TSTRIDE + 0];
  const float ps_si = W[si * TSTRIDE + si];
  const float ps_sj = W[sj * TSTRIDE + si];
  const float cur = (1.0f / 16.0f) *
      (p0_si * p0_sj + ((si == sj && si != 0) ? ps_si * ps_sj : 0.0f));
  const float term = (c != i) ? logf(cur) : 0.0f;

  if (lane == 0) wsum[wave] = term;
  __syncthreads();
  if (threadIdx.x == 0) {
    float b = 0.0f;
#pragma unroll
    for (int w = 0; w < WAVES_PER_BLOCK; ++w) b += wsum[w];
    partial[blockIdx.x] = b;
  }
}

__global__ __launch_bounds__(256) void logalike_reduce_kernel(
    const float* __restrict__ partial, int n, float* __restrict__ out) {
  __shared__ float sm[256];
  float s = 0.0f;
  for (int idx = threadIdx.x; idx < n; idx += 256) s += partial[idx];
  sm[threadIdx.x] = s;
  __syncthreads();
  for (int off = 128; off > 0; off >>= 1) {
    if (threadIdx.x < off) sm[threadIdx.x] += sm[threadIdx.x + off];
    __syncthreads();
  }
  if (threadIdx.x == 0) out[0] = sm[0];
}

extern "C" void kernel_launch(void* const* d_in, const int* in_sizes, int n_in,
                              void* d_out, int out_size, void* d_ws, size_t ws_size,
                              hipStream_t stream) {
  const float* X  = (const float*)d_in[0];   // (512, 9)
  const float* Q  = (const float*)d_in[1];   // (256, 16, 16)
  const int*   ch = (const int*)d_in[2];     // (512, 256)
  const int*   ip = (const int*)d_in[3];     // scalar i
  float* out      = (float*)d_out;           // 1 float
  float* partial  = (float*)d_ws;            // 16384 floats of scratch

  const int nblocks = (C_N * S_N) / WAVES_PER_BLOCK;   // 16384
  logalike_expm_kernel<<<nblocks, 256, 0, stream>>>(X, Q, ch, ip, partial);
  logalike_reduce_kernel<<<1, 256, 0, stream>>>(partial, nblocks, out);
}